// LTC_22505628631562
// MI455X (gfx1250) — compile-verified
//
#include <hip/hip_runtime.h>
#include <stdint.h>

#define B_   128
#define T_   64
#define I_   128
#define H_   256
#define UNF  6
#define EPS_ 1e-8f
#define NB   4                 // batches per workgroup
#define QS   4                 // i-dimension split
#define NTHR (H_ * QS)         // 1024 threads = 32 waves

// ---------------------------------------------------------------------------
// sigmoid(x) = 0.5 + 0.5*tanh(x/2); gfx1250 has v_tanh_f32 (1 trans op).
static __device__ __forceinline__ float fast_tanhf(float a) {
#if __has_builtin(__builtin_amdgcn_tanhf)
  return __builtin_amdgcn_tanhf(a);
#else
  // tanh(a) = 2/(1+exp2(-2a*log2e)) - 1
  float e = __builtin_amdgcn_exp2f(-2.885390081777927f * a);
  return fmaf(2.0f, __builtin_amdgcn_rcpf(1.0f + e), -1.0f);
#endif
}

#if defined(__AMDGCN__) && __has_builtin(__builtin_amdgcn_global_load_async_to_lds_b32)
#define HAVE_ASYNC_LDS 1
typedef __attribute__((address_space(1))) int* as1_i32p;
typedef __attribute__((address_space(3))) int* as3_i32p;
#endif

static __device__ __forceinline__ void wait_async0() {
#if __has_builtin(__builtin_amdgcn_s_wait_asynccnt)
  __builtin_amdgcn_s_wait_asynccnt(0);
#else
  asm volatile("s_wait_asynccnt 0" ::: "memory");
#endif
}

// ---------------------------------------------------------------------------
// Main LTC kernel first in the file so the disasm snippet shows it.
// Thread = (i-quarter q, output neuron j); NB batches per workgroup.
// V state lives in LDS as [i][b] float4 rows (wave32 broadcast reads).
__global__ __launch_bounds__(NTHR, 1)
void ltc_main(const float* __restrict__ x,
              const float4* __restrict__ pk_sen,   // [I_*H_]
              const float4* __restrict__ pk_int,   // [H_*H_]
              const float* __restrict__ cns, const float* __restrict__ cds,
              const float* __restrict__ cni, const float* __restrict__ cdi,
              const float* __restrict__ glvl, const float* __restrict__ cmt,
              const float* __restrict__ denb,
              const float* __restrict__ ow, const float* __restrict__ ob,
              float* __restrict__ out, float* __restrict__ hT) {
  __shared__ __align__(16) float Vt[H_][NB];        // V state, [i][b]
  __shared__ __align__(16) float xs[I_][NB];        // staged x tile, [i][b]
  __shared__ float pn[QS][NB][H_];                  // partial numerators
  __shared__ float pd[QS][NB][H_];                  // partial denominators

  const int tid = threadIdx.x;
  const int j   = tid & (H_ - 1);   // output neuron
  const int q   = tid >> 8;         // i-quarter
  const int gb  = blockIdx.x * NB;  // first batch of this workgroup

  // per-j loop invariants, hoisted out of the t/unfold loops
  const float csn = cns[j], csd = cds[j];
  const float cjn = cni[j], cjd = cdi[j];
  const float cmj = cmt[j], gl  = glvl[j], db = denb[j];
  const float wo  = ow[j],  oo  = ob[j];

  if (q == 0) {
#pragma unroll
    for (int b = 0; b < NB; ++b) Vt[j][b] = 0.f;
  }
  __syncthreads();

  for (int t = 0; t < T_; ++t) {
    // ---- stage x[gb..gb+NB, t, :] into LDS (async DMA path on gfx1250) ----
#ifdef HAVE_ASYNC_LDS
    if (tid < NB * I_) {
      const int i = tid & (I_ - 1);
      const int b = tid >> 7;
      const float* src = &x[((size_t)(gb + b) * T_ + t) * I_ + i];
      __builtin_amdgcn_global_load_async_to_lds_b32(
          (as1_i32p)(uintptr_t)src,
          (as3_i32p)(uintptr_t)&xs[i][b],
          0, 0);
    }
    wait_async0();
#else
    if (tid < NB * I_) {
      const int i = tid & (I_ - 1);
      const int b = tid >> 7;
      xs[i][b] = x[((size_t)(gb + b) * T_ + t) * I_ + i];
    }
#endif
    __syncthreads();

    // ---- sensing partial sums over this thread's i-quarter ----
    float sn[NB], sd[NB];
#pragma unroll
    for (int b = 0; b < NB; ++b) { sn[b] = 0.f; sd[b] = 0.f; }
    {
      const int i0 = q * (I_ / QS);
      const float4* pp = pk_sen + (size_t)i0 * H_ + j;
#pragma unroll 4
      for (int i = 0; i < I_ / QS; ++i) {
        float4 p = pp[(size_t)i * H_];
        float4 xv = *reinterpret_cast<const float4*>(xs[i0 + i]);
        const float* xa = reinterpret_cast<const float*>(&xv);
#pragma unroll
        for (int b = 0; b < NB; ++b) {
          float tt = fast_tanhf(fmaf(p.x, xa[b], p.y));
          sn[b] = fmaf(p.z, tt, sn[b]);
          sd[b] = fmaf(p.w, tt, sd[b]);
        }
      }
    }
#pragma unroll
    for (int b = 0; b < NB; ++b) { pn[q][b][j] = sn[b]; pd[q][b][j] = sd[b]; }
    __syncthreads();

    float rns[NB], rds[NB];           // sensing totals (valid on q==0 only)
#pragma unroll
    for (int b = 0; b < NB; ++b) { rns[b] = 0.f; rds[b] = 0.f; }
    if (q == 0) {
#pragma unroll
      for (int b = 0; b < NB; ++b) {
        rns[b] = pn[0][b][j] + pn[1][b][j] + pn[2][b][j] + pn[3][b][j] + csn;
        rds[b] = pd[0][b][j] + pd[1][b][j] + pd[2][b][j] + pd[3][b][j] + csd;
      }
    }
    __syncthreads();   // protect pn/pd before unfolds overwrite them

    // ---- 6 unfolds ----
    for (int u = 0; u < UNF; ++u) {
      float an[NB], ad[NB];
#pragma unroll
      for (int b = 0; b < NB; ++b) { an[b] = 0.f; ad[b] = 0.f; }
      const int i0 = q * (H_ / QS);
      const float4* pp = pk_int + (size_t)i0 * H_ + j;
#pragma unroll 4
      for (int i = 0; i < H_ / QS; ++i) {
        float4 p = pp[(size_t)i * H_];
        float4 vv = *reinterpret_cast<const float4*>(Vt[i0 + i]);
        const float* va = reinterpret_cast<const float*>(&vv);
#pragma unroll
        for (int b = 0; b < NB; ++b) {
          float tt = fast_tanhf(fmaf(p.x, va[b], p.y));
          an[b] = fmaf(p.z, tt, an[b]);
          ad[b] = fmaf(p.w, tt, ad[b]);
        }
      }
#pragma unroll
      for (int b = 0; b < NB; ++b) { pn[q][b][j] = an[b]; pd[q][b][j] = ad[b]; }
      __syncthreads();
      if (q == 0) {
        float4 vnew;
        float* vp = reinterpret_cast<float*>(&vnew);
#pragma unroll
        for (int b = 0; b < NB; ++b) {
          float wn = pn[0][b][j] + pn[1][b][j] + pn[2][b][j] + pn[3][b][j] + cjn + rns[b];
          float wd = pd[0][b][j] + pd[1][b][j] + pd[2][b][j] + pd[3][b][j] + cjd + rds[b];
          float vo = Vt[j][b];
          vp[b] = (vo * cmj + gl + wn) / (db + wd);
        }
        *reinterpret_cast<float4*>(Vt[j]) = vnew;
      }
      __syncthreads();
    }

    // ---- output for this timestep ----
    if (q == 0) {
#pragma unroll
      for (int b = 0; b < NB; ++b)
        out[((size_t)(gb + b) * T_ + t) * H_ + j] = fmaf(Vt[j][b], wo, oo);
    }
    // racing threads block at next timestep's first barrier until q==0 is done
  }

  if (q == 0) {
#pragma unroll
    for (int b = 0; b < NB; ++b)
      hT[(size_t)(gb + b) * H_ + j] = Vt[j][b];
  }
}

// ---------------------------------------------------------------------------
// Pack params into float4: (x=0.5*sigma*scale, y=0.5*sigma*(bias-mu),
//                           z=0.5*w*erev,      w=0.5*w)
// For sensing, fold the input affine (iw,ib) into x,y. rows = I_ or H_.
__global__ void prep_pack(const float* __restrict__ mu, const float* __restrict__ sg,
                          const float* __restrict__ w,  const float* __restrict__ er,
                          const float* __restrict__ iw, const float* __restrict__ ib,
                          float4* __restrict__ pk, int rows, int useAffine) {
  int idx = blockIdx.x * blockDim.x + threadIdx.x;
  if (idx >= rows * H_) return;
  int i = idx / H_;
  float s = 0.5f * sg[idx];
  float m = mu[idx];
  float wv = 0.5f * w[idx];
  float4 p;
  if (useAffine) { p.x = s * iw[i]; p.y = s * (ib[i] - m); }
  else           { p.x = s;         p.y = -s * m; }
  p.z = wv * er[idx];
  p.w = wv;
  pk[idx] = p;
}

// Per-output-neuron constants: tanh-identity offsets + leak/cm terms.
__global__ void prep_consts(const float* __restrict__ sw, const float* __restrict__ ser,
                            const float* __restrict__ iw, const float* __restrict__ ier,
                            const float* __restrict__ vleak, const float* __restrict__ gleak,
                            const float* __restrict__ cm,
                            float* __restrict__ cns, float* __restrict__ cds,
                            float* __restrict__ cni, float* __restrict__ cdi,
                            float* __restrict__ glvl, float* __restrict__ cmt,
                            float* __restrict__ denb) {
  int j = blockIdx.x * blockDim.x + threadIdx.x;
  if (j >= H_) return;
  float ns = 0.f, ds = 0.f, ni = 0.f, di = 0.f;
  for (int i = 0; i < I_; ++i) {
    float wv = 0.5f * sw[i * H_ + j];
    ns += wv * ser[i * H_ + j];
    ds += wv;
  }
  for (int i = 0; i < H_; ++i) {
    float wv = 0.5f * iw[i * H_ + j];
    ni += wv * ier[i * H_ + j];
    di += wv;
  }
  cns[j] = ns; cds[j] = ds; cni[j] = ni; cdi[j] = di;
  float g = gleak[j];
  glvl[j] = g * vleak[j];
  float c = cm[j] * (float)UNF;
  cmt[j]  = c;
  denb[j] = c + g + EPS_;
}

// ---------------------------------------------------------------------------
extern "C" void kernel_launch(void* const* d_in, const int* in_sizes, int n_in,
                              void* d_out, int out_size, void* d_ws, size_t ws_size,
                              hipStream_t stream) {
  (void)in_sizes; (void)n_in; (void)out_size; (void)ws_size;

  const float* x     = (const float*)d_in[0];
  const float* iw    = (const float*)d_in[1];
  const float* ibv   = (const float*)d_in[2];
  const float* smu   = (const float*)d_in[3];
  const float* ssg   = (const float*)d_in[4];
  const float* sw    = (const float*)d_in[5];
  const float* ser   = (const float*)d_in[6];
  const float* imu   = (const float*)d_in[7];
  const float* isg   = (const float*)d_in[8];
  const float* iww   = (const float*)d_in[9];
  const float* ier   = (const float*)d_in[10];
  const float* vleak = (const float*)d_in[11];
  const float* gleak = (const float*)d_in[12];
  const float* cm    = (const float*)d_in[13];
  const float* ow    = (const float*)d_in[14];
  const float* ob    = (const float*)d_in[15];

  float* out = (float*)d_out;
  float* hT  = out + (size_t)B_ * T_ * H_;

  // workspace carve-up
  float4* pk_sen = (float4*)d_ws;                       // I_*H_ float4
  float4* pk_int = pk_sen + (size_t)I_ * H_;            // H_*H_ float4
  float*  fbase  = (float*)(pk_int + (size_t)H_ * H_);
  float* cns  = fbase + 0 * H_;
  float* cds  = fbase + 1 * H_;
  float* cni  = fbase + 2 * H_;
  float* cdi  = fbase + 3 * H_;
  float* glvl = fbase + 4 * H_;
  float* cmt  = fbase + 5 * H_;
  float* denb = fbase + 6 * H_;

  prep_pack<<<(I_ * H_ + 255) / 256, 256, 0, stream>>>(smu, ssg, sw, ser, iw, ibv,
                                                       pk_sen, I_, 1);
  prep_pack<<<(H_ * H_ + 255) / 256, 256, 0, stream>>>(imu, isg, iww, ier, nullptr, nullptr,
                                                       pk_int, H_, 0);
  prep_consts<<<1, H_, 0, stream>>>(sw, ser, iww, ier, vleak, gleak, cm,
                                    cns, cds, cni, cdi, glvl, cmt, denb);

  ltc_main<<<B_ / NB, NTHR, 0, stream>>>(x, pk_sen, pk_int,
                                         cns, cds, cni, cdi, glvl, cmt, denb,
                                         ow, ob, out, hT);
}